// GraphUNet_33543694582268
// MI455X (gfx1250) — compile-verified
//
#include <hip/hip_runtime.h>
#include <hip/hip_bf16.h>

#define NN 50000
#define NE 800000
#define EPSF 1e-5f

typedef __attribute__((ext_vector_type(2))) float v2f;
typedef __attribute__((ext_vector_type(8))) float v8f;

// ---------------- GEMM: C[n,dout] = X[n,K] @ W[K,dout], f32 WMMA 16x16x4 ----
// one wave = one 16x16 output tile. n multiple of 16, K multiple of 4,
// dout multiple of 16. ldx/ldc allow strided (concat) feature buffers.
__global__ __launch_bounds__(32) void k_gemm16(
    float* __restrict__ C, int ldc,
    const float* __restrict__ X, int ldx,
    const float* __restrict__ W, int K, int dout)
{
    const int lane = threadIdx.x;
    const int half = lane >> 4;        // 0: lanes 0-15, 1: lanes 16-31
    const int nl   = lane & 15;
    const int row0 = blockIdx.x << 4;
    const int col  = (blockIdx.y << 4) + nl;

    const float* xr = X + (size_t)(row0 + nl) * ldx + (half << 1);
    const float* wc = W + (size_t)(half << 1) * dout + col;

    v8f acc = {0.f, 0.f, 0.f, 0.f, 0.f, 0.f, 0.f, 0.f};
    for (int k = 0; k < K; k += 4) {
        v2f a, b;
        a[0] = xr[0];            // A: 16x4 tile, K = 2*half + j
        a[1] = xr[1];
        b[0] = wc[0];            // B: 4x16 tile, same K striping
        b[1] = wc[dout];
        acc = __builtin_amdgcn_wmma_f32_16x16x4_f32(
            false, a, false, b, (short)0, acc, false, false);
        xr += 4;
        wc += (size_t)dout * 4;
    }
    // D layout: VGPR j -> row = row0 + j + 8*half, col = col0 + nl
    float* cr = C + (size_t)(row0 + (half << 3)) * ldc + col;
#pragma unroll
    for (int j = 0; j < 8; ++j) cr[(size_t)j * ldc] = acc[j];
}

// ---------------- normalization precompute ---------------------------------
__global__ void k_fill(float* __restrict__ p, float v, int n) {
    int i = blockIdx.x * blockDim.x + threadIdx.x;
    if (i < n) p[i] = v;
}
__global__ void k_deg(const int* __restrict__ dst, float* __restrict__ deg, int e) {
    int i = blockIdx.x * blockDim.x + threadIdx.x;
    if (i < e) atomicAdd(&deg[dst[i]], 1.0f);
}
__global__ void k_rsqrt(float* __restrict__ d, int n) {
    int i = blockIdx.x * blockDim.x + threadIdx.x;
    if (i < n) d[i] = rsqrtf(d[i]);          // deg >= 1 (self loop)
}
__global__ void k_edgew(const int* __restrict__ src, const int* __restrict__ dst,
                        const float* __restrict__ dis, float* __restrict__ ew, int e) {
    int i = blockIdx.x * blockDim.x + threadIdx.x;
    if (i < e) ew[i] = dis[src[i]] * dis[dst[i]];
}

// ---------------- aggregation ----------------------------------------------
// acc[i,c] = h[i,c]*dis[i]^2 + bias[c]   (self-loop + bias init)
__global__ void k_init_agg(float* __restrict__ acc, const float* __restrict__ h,
                           const float* __restrict__ bias, const float* __restrict__ dis,
                           long long total, int shift, int mask)
{
    long long idx = (long long)blockIdx.x * blockDim.x + threadIdx.x;
    if (idx >= total) return;
    int i = (int)(idx >> shift), c = (int)idx & mask;
    float ds = dis[i];
    acc[idx] = h[idx] * ds * ds + bias[c];
}
// acc[dst[e],c] += h[src[e],c] * ew[e]   (flat over edge x channel)
__global__ void k_scatter(float* __restrict__ acc, const float* __restrict__ h,
                          const int* __restrict__ src, const int* __restrict__ dst,
                          const float* __restrict__ ew, long long total, int shift, int mask)
{
    long long idx = (long long)blockIdx.x * blockDim.x + threadIdx.x;
    if (idx >= total) return;
    int e = (int)(idx >> shift), c = (int)idx & mask;
    int s = src[e], d = dst[e];
    float v = h[((size_t)s << shift) + c] * ew[e];
    atomicAdd(&acc[((size_t)d << shift) + c], v);
}

// ---------------- batchnorm ------------------------------------------------
__global__ void k_zero_stats(float* __restrict__ s) { s[threadIdx.x] = 0.0f; }

// blockDim.x == dout; each block reduces a row-chunk, coalesced along columns
__global__ void k_bn_stats(const float* __restrict__ acc, float* __restrict__ stats,
                           int n, int dout, int rows_per_block)
{
    int c  = threadIdx.x;
    int r0 = blockIdx.x * rows_per_block;
    int r1 = r0 + rows_per_block; if (r1 > n) r1 = n;
    float s = 0.f, s2 = 0.f;
    for (int i = r0; i < r1; ++i) {
        float v = acc[(size_t)i * dout + c];
        s += v; s2 += v * v;
    }
    atomicAdd(&stats[c], s);
    atomicAdd(&stats[dout + c], s2);
}

// y = relu(g*(v-m)*rsqrt(var+eps)+be)   (or plain relu if g == nullptr)
__global__ void k_bn_apply(float* __restrict__ out, int ldo,
                           const float* __restrict__ acc, const float* __restrict__ stats,
                           const float* __restrict__ g, const float* __restrict__ be,
                           long long total, int shift, int mask, float invn)
{
    long long idx = (long long)blockIdx.x * blockDim.x + threadIdx.x;
    if (idx >= total) return;
    int i = (int)(idx >> shift), c = (int)idx & mask;
    float v = acc[idx];
    if (g) {
        float m   = stats[c] * invn;
        float var = stats[(mask + 1) + c] * invn - m * m;
        v = g[c] * (v - m) * rsqrtf(var + EPSF) + be[c];
    }
    out[(size_t)i * ldo + c] = fmaxf(v, 0.0f);
}

// buf[i,c] += b[c]; optional relu (for FC layers)
__global__ void k_bias(float* __restrict__ buf, const float* __restrict__ b,
                       long long total, int mask, int relu)
{
    long long idx = (long long)blockIdx.x * blockDim.x + threadIdx.x;
    if (idx >= total) return;
    float v = buf[idx] + b[(int)idx & mask];
    buf[idx] = relu ? fmaxf(v, 0.0f) : v;
}

// ---------------- host side ------------------------------------------------
static inline unsigned nblk(long long total, int b) { return (unsigned)((total + b - 1) / b); }

static void run_gcn_block(const float* Xin, int ldx, int K, int dout,
                          const float* W, const float* b, const float* g, const float* be,
                          float* Yout, int ldo,
                          float* Th, float* Ta, float* stats,
                          const float* dis, const int* src, const int* dst, const float* ew,
                          hipStream_t stream)
{
    const int n = NN, blk = 256;
    const int shift = __builtin_ctz(dout), mask = dout - 1;
    dim3 gg(n / 16, dout / 16);
    k_gemm16<<<gg, 32, 0, stream>>>(Th, dout, Xin, ldx, W, K, dout);

    long long ntot = (long long)n * dout;
    k_init_agg<<<nblk(ntot, blk), blk, 0, stream>>>(Ta, Th, b, dis, ntot, shift, mask);

    long long etot = (long long)NE * dout;
    k_scatter<<<nblk(etot, blk), blk, 0, stream>>>(Ta, Th, src, dst, ew, etot, shift, mask);

    if (g) {
        k_zero_stats<<<1, 2 * dout, 0, stream>>>(stats);
        const int rpb = 128;
        k_bn_stats<<<nblk(n, rpb), dout, 0, stream>>>(Ta, stats, n, dout, rpb);
    }
    k_bn_apply<<<nblk(ntot, blk), blk, 0, stream>>>(Yout, ldo, Ta, stats, g, be,
                                                    ntot, shift, mask, 1.0f / n);
}

extern "C" void kernel_launch(void* const* d_in, const int* in_sizes, int n_in,
                              void* d_out, int out_size, void* d_ws, size_t ws_size,
                              hipStream_t stream)
{
    const float* x   = (const float*)d_in[0];
    const int*   ei  = (const int*)d_in[1];
    const int*   src = ei;
    const int*   dst = ei + NE;

    const float *w_in = (const float*)d_in[2],  *b_in = (const float*)d_in[3],
                *g_in = (const float*)d_in[4],  *be_in = (const float*)d_in[5];
    const float *w_d1 = (const float*)d_in[6],  *b_d1 = (const float*)d_in[7],
                *g_d1 = (const float*)d_in[8],  *be_d1 = (const float*)d_in[9];
    const float *w_d2 = (const float*)d_in[10], *b_d2 = (const float*)d_in[11],
                *g_d2 = (const float*)d_in[12], *be_d2 = (const float*)d_in[13];
    const float *w_d3 = (const float*)d_in[14], *b_d3 = (const float*)d_in[15],
                *g_d3 = (const float*)d_in[16], *be_d3 = (const float*)d_in[17];
    const float *w_lat = (const float*)d_in[18], *b_lat = (const float*)d_in[19],
                *g_lat = (const float*)d_in[20], *be_lat = (const float*)d_in[21];
    const float *w_u3 = (const float*)d_in[22], *b_u3 = (const float*)d_in[23],
                *g_u3 = (const float*)d_in[24], *be_u3 = (const float*)d_in[25];
    const float *w_u2 = (const float*)d_in[26], *b_u2 = (const float*)d_in[27],
                *g_u2 = (const float*)d_in[28], *be_u2 = (const float*)d_in[29];
    const float *w_u1 = (const float*)d_in[30], *b_u1 = (const float*)d_in[31];
    const float *w_o1 = (const float*)d_in[32], *b_o1 = (const float*)d_in[33],
                *g_o1 = (const float*)d_in[34], *be_o1 = (const float*)d_in[35];
    const float *w_fc1 = (const float*)d_in[36], *b_fc1 = (const float*)d_in[37];
    const float *w_fc2 = (const float*)d_in[38], *b_fc2 = (const float*)d_in[39];

    // workspace bump allocator
    float* ws = (float*)d_ws;
    float* dis   = ws; ws += NN;                 // deg then rsqrt in place
    float* ew    = ws; ws += NE;
    float* Th    = ws; ws += (size_t)NN * 256;   // GEMM result
    float* Ta    = ws; ws += (size_t)NN * 256;   // aggregated
    float* stats = ws; ws += 512;
    float* F0    = ws; ws += (size_t)NN * 64;    // [x7 | x0]
    float* F1    = ws; ws += (size_t)NN * 128;   // [x6 | x1]
    float* F2    = ws; ws += (size_t)NN * 256;   // [x5 | x2]
    float* F3    = ws; ws += (size_t)NN * 512;   // [x4 | x3]
    float* X8    = ws; ws += (size_t)NN * 32;    // o1 block output

    // ---- GCN symmetric normalization: D^-1/2 (A+I) D^-1/2 ----
    const int blk = 256;
    k_fill<<<nblk(NN, blk), blk, 0, stream>>>(dis, 1.0f, NN);        // self loop
    k_deg<<<nblk(NE, blk), blk, 0, stream>>>(dst, dis, NE);
    k_rsqrt<<<nblk(NN, blk), blk, 0, stream>>>(dis, NN);
    k_edgew<<<nblk(NE, blk), blk, 0, stream>>>(src, dst, dis, ew, NE);

    // ---- encoder ----
    run_gcn_block(x,        64,  64,  32, w_in,  b_in,  g_in,  be_in,  F0 + 32, 64,  Th, Ta, stats, dis, src, dst, ew, stream);
    run_gcn_block(F0 + 32,  64,  32,  64, w_d1,  b_d1,  g_d1,  be_d1,  F1 + 64, 128, Th, Ta, stats, dis, src, dst, ew, stream);
    run_gcn_block(F1 + 64,  128, 64,  128, w_d2, b_d2,  g_d2,  be_d2,  F2 + 128, 256, Th, Ta, stats, dis, src, dst, ew, stream);
    run_gcn_block(F2 + 128, 256, 128, 256, w_d3, b_d3,  g_d3,  be_d3,  F3 + 256, 512, Th, Ta, stats, dis, src, dst, ew, stream);
    run_gcn_block(F3 + 256, 512, 256, 256, w_lat, b_lat, g_lat, be_lat, F3,      512, Th, Ta, stats, dis, src, dst, ew, stream);
    // ---- decoder (concat via strided buffers) ----
    run_gcn_block(F3, 512, 512, 128, w_u3, b_u3, g_u3, be_u3, F2, 256, Th, Ta, stats, dis, src, dst, ew, stream);
    run_gcn_block(F2, 256, 256, 64,  w_u2, b_u2, g_u2, be_u2, F1, 128, Th, Ta, stats, dis, src, dst, ew, stream);
    run_gcn_block(F1, 128, 128, 32,  w_u1, b_u1, nullptr, nullptr, F0, 64, Th, Ta, stats, dis, src, dst, ew, stream); // no BN
    run_gcn_block(F0, 64,  64,  32,  w_o1, b_o1, g_o1, be_o1, X8, 32, Th, Ta, stats, dis, src, dst, ew, stream);

    // ---- FC head ----
    {
        dim3 g1(NN / 16, 32 / 16);
        k_gemm16<<<g1, 32, 0, stream>>>(Th, 32, X8, 32, w_fc1, 32, 32);
        long long t1 = (long long)NN * 32;
        k_bias<<<nblk(t1, blk), blk, 0, stream>>>(Th, b_fc1, t1, 31, 1);

        dim3 g2(NN / 16, 16 / 16);
        float* out = (float*)d_out;
        k_gemm16<<<g2, 32, 0, stream>>>(out, 16, Th, 32, w_fc2, 32, 16);
        long long t2 = (long long)NN * 16;
        k_bias<<<nblk(t2, blk), blk, 0, stream>>>(out, b_fc2, t2, 15, 0);
    }
}